// SpectralConv2d_86612310491734
// MI455X (gfx1250) — compile-verified
//
#include <hip/hip_runtime.h>

typedef __attribute__((ext_vector_type(2))) float v2f;
typedef __attribute__((ext_vector_type(8))) float v8f;

// ---------------------------------------------------------------------------
// WMMA helpers: V_WMMA_F32_16X16X4_F32  (D(16x16,f32) = A(16x4,f32)*B(4x16,f32)+C)
// A frag (16x4): lane L holds M=L&15; VGPR v = A[M][k0 + ((L>=16)?2:0) + v]
// B frag (4x16): lane L holds N=L&15; VGPR v = B[k0 + ((L>=16)?2:0) + v][N]
// C/D frag: VGPR r: lanes 0-15 -> M=r, lanes 16-31 -> M=r+8; N = L&15
// ---------------------------------------------------------------------------
static __device__ __forceinline__ v8f wmma4(v2f a, v2f b, v8f c) {
  return __builtin_amdgcn_wmma_f32_16x16x4_f32(false, a, false, b, (short)0, c,
                                               false, false);
}

static __device__ __forceinline__ v2f frag_a(const float* A, int lda, int m0,
                                             int k0, int lane) {
  const float* p = A + (m0 + (lane & 15)) * lda + k0 + ((lane >> 4) << 1);
  v2f r = {p[0], p[1]};
  return r;
}
static __device__ __forceinline__ v2f frag_b(const float* B, int ldb, int k0,
                                             int n0, int lane) {
  const float* p = B + (k0 + ((lane >> 4) << 1)) * ldb + n0 + (lane & 15);
  v2f r = {p[0], p[ldb]};
  return r;
}
// A read from transposed storage: A[m][k] = T[k*ldt + m]
static __device__ __forceinline__ v2f frag_at(const float* T, int ldt, int m0,
                                              int k0, int lane) {
  const float* p = T + (k0 + ((lane >> 4) << 1)) * ldt + m0 + (lane & 15);
  v2f r = {p[0], p[ldt]};
  return r;
}
// B read from transposed storage: B[k][n] = S[n*lds + k]
static __device__ __forceinline__ v2f frag_bt(const float* S, int lds, int k0,
                                              int n0, int lane) {
  const float* p = S + (n0 + (lane & 15)) * lds + k0 + ((lane >> 4) << 1);
  v2f r = {p[0], p[1]};
  return r;
}
static __device__ __forceinline__ void store_d(float* D, int ldd, int m0,
                                               int n0, int lane, v8f acc) {
  int m = m0 + ((lane >> 4) << 3);
  int n = n0 + (lane & 15);
#pragma unroll
  for (int r = 0; r < 8; ++r) D[(m + r) * ldd + n] = acc[r];
}

// ---------------------------------------------------------------------------
// Kernel 0: build the 128x128 pseudo-Hartley matrix of the reference's
// recursive fht1d.  Telescoped closed form of the recursion:
//   M[k,n] = prod over bits j of n:  bit set ->  s * cas(2*pi*kj/Nj)
//   with Nj = 128>>j, kj = k mod (Nj/2), s = -1 iff bit (6-j) of k set.
// ---------------------------------------------------------------------------
__global__ void build_M_kernel(float* __restrict__ M) {
  int idx = blockIdx.x * blockDim.x + threadIdx.x;  // 0..16383
  int k = idx >> 7, n = idx & 127;
  float C = 1.0f;
#pragma unroll
  for (int j = 0; j < 7; ++j) {
    if ((n >> j) & 1) {
      int Nj = 128 >> j;
      int half = Nj >> 1;
      int kj = k & (half - 1);
      float ang = 6.28318530717958647692f * (float)kj / (float)Nj;
      float c = __cosf(ang) + __sinf(ang);
      if ((k >> (6 - j)) & 1) c = -c;
      C *= c;
    }
  }
  M[k * 128 + n] = C;
}

// ---------------------------------------------------------------------------
// Kernel 1: forward truncated transform, one workgroup per (b, cin).
//   T(32x128)  = M_top(32x128) * x(128x128)      (x streamed from global)
//   Xt(32x32)  = T * M_top^T
// Xt stored mode-major: Xt[(k*32+l)*2048 + (b*64+c)]
// ---------------------------------------------------------------------------
__global__ void __launch_bounds__(128) fwd_kernel(const float* __restrict__ x,
                                                  const float* __restrict__ M,
                                                  float* __restrict__ Xt) {
  __shared__ float lM[32 * 128];  // 16 KB
  __shared__ float lT[32 * 128];  // 16 KB
  const int tid = threadIdx.x, lane = tid & 31, wave = tid >> 5;
  const int bc = blockIdx.x;
  const float* xg = x + (size_t)bc * 16384;

  for (int i = tid; i < 1024; i += 128)
    ((float4*)lM)[i] = ((const float4*)M)[i];  // rows 0..31 of M
  __syncthreads();

  // --- T = M_top * x : tiles mt in {0,1}, wave owns nt = 2*wave + {0,1}
  {
    const int n0 = wave * 32, n1 = n0 + 16;
    v8f a00 = {}, a01 = {}, a10 = {}, a11 = {};
    for (int k0 = 0; k0 < 128; k0 += 4) {
      v2f A0 = frag_a(lM, 128, 0, k0, lane);
      v2f A1 = frag_a(lM, 128, 16, k0, lane);
      v2f B0 = frag_b(xg, 128, k0, n0, lane);  // x read once, straight from HBM
      v2f B1 = frag_b(xg, 128, k0, n1, lane);
      a00 = wmma4(A0, B0, a00);
      a01 = wmma4(A0, B1, a01);
      a10 = wmma4(A1, B0, a10);
      a11 = wmma4(A1, B1, a11);
    }
    store_d(lT, 128, 0, n0, lane, a00);
    store_d(lT, 128, 0, n1, lane, a01);
    store_d(lT, 128, 16, n0, lane, a10);
    store_d(lT, 128, 16, n1, lane, a11);
  }
  __syncthreads();

  // --- Xt = T * M_top^T : 4 tiles, one per wave
  {
    const int m0 = (wave >> 1) * 16, n0 = (wave & 1) * 16;
    v8f acc = {};
    for (int k0 = 0; k0 < 128; k0 += 4) {
      v2f A = frag_a(lT, 128, m0, k0, lane);
      v2f B = frag_bt(lM, 128, k0, n0, lane);  // B[w][l] = M[l][w]
      acc = wmma4(A, B, acc);
    }
    const int kb = m0 + ((lane >> 4) << 3);
    const int l = n0 + (lane & 15);
#pragma unroll
    for (int r = 0; r < 8; ++r)
      Xt[(size_t)((kb + r) * 32 + l) * 2048 + bc] = acc[r];
  }
}

// ---------------------------------------------------------------------------
// Kernel 2: per-mode channel mixing, one workgroup per mode (k,l).
//   Out(b,o) = 0.5 * ( X[mode](b,i)*Wp(i,o) + X[negmode](b,i)*Wm(i,o) )
//   Wp = w(k,l)+w(-k,-l), Wm = w(k,l)-w(-k,-l)   (built on the fly; weights in L2)
// Blk stored as [(b*64+o)*1024 + mode] for the inverse stage.
// ---------------------------------------------------------------------------
__global__ void __launch_bounds__(256) mix_kernel(const float* __restrict__ Xt,
                                                  const float* __restrict__ W,
                                                  float* __restrict__ Blk) {
  __shared__ float lX[2048];   // [b][i] ld 64
  __shared__ float lXn[2048];  // neg-freq partner
  __shared__ float lWp[4096];  // [i][o] ld 64
  __shared__ float lWm[4096];
  const int tid = threadIdx.x, lane = tid & 31, wave = tid >> 5;
  const int mode = blockIdx.x;
  const int k = mode >> 5, l = mode & 31;
  const int negmode = (((32 - k) & 31) << 5) + ((32 - l) & 31);

  for (int i = tid; i < 2048; i += 256) {
    lX[i] = Xt[(size_t)mode * 2048 + i];
    lXn[i] = Xt[(size_t)negmode * 2048 + i];
  }
  for (int i = tid; i < 4096; i += 256) {  // i = ci*64 + co
    float w1 = W[(size_t)i * 1024 + mode];
    float w2 = W[(size_t)i * 1024 + negmode];
    lWp[i] = w1 + w2;
    lWm[i] = w1 - w2;
  }
  __syncthreads();

  // D(32x64): 8 tiles over 8 waves
  const int m0 = (wave >> 2) * 16, n0 = (wave & 3) * 16;
  v8f acc = {};
  for (int k0 = 0; k0 < 64; k0 += 4) {
    v2f A = frag_a(lX, 64, m0, k0, lane);
    v2f B = frag_b(lWp, 64, k0, n0, lane);
    acc = wmma4(A, B, acc);
  }
  for (int k0 = 0; k0 < 64; k0 += 4) {
    v2f A = frag_a(lXn, 64, m0, k0, lane);
    v2f B = frag_b(lWm, 64, k0, n0, lane);
    acc = wmma4(A, B, acc);
  }
  const int b0 = m0 + ((lane >> 4) << 3);
  const int o = n0 + (lane & 15);
#pragma unroll
  for (int r = 0; r < 8; ++r)
    Blk[(size_t)((b0 + r) * 64 + o) * 1024 + mode] = 0.5f * acc[r];
}

// ---------------------------------------------------------------------------
// Kernel 3: inverse transform, one workgroup per (b, cout).
//   U(128x32) = M_top^T * Blk(32x32)
//   y(128x128) = U * M_top / 16384
// ---------------------------------------------------------------------------
__global__ void __launch_bounds__(128) inv_kernel(const float* __restrict__ Blk,
                                                  const float* __restrict__ M,
                                                  float* __restrict__ out) {
  __shared__ float lM[32 * 128];  // 16 KB
  __shared__ float lB[32 * 32];   // 4 KB
  __shared__ float lU[128 * 32];  // 16 KB
  const int tid = threadIdx.x, lane = tid & 31, wave = tid >> 5;
  const int bo = blockIdx.x;

  for (int i = tid; i < 1024; i += 128)
    ((float4*)lM)[i] = ((const float4*)M)[i];
  for (int i = tid; i < 256; i += 128)
    ((float4*)lB)[i] = ((const float4*)(Blk + (size_t)bo * 1024))[i];
  __syncthreads();

  // --- U = G * Blk, G[h][k] = M[k][h]; wave owns rows m0..m0+31
  {
    const int m0 = wave * 32;
    v8f a00 = {}, a01 = {}, a10 = {}, a11 = {};
    for (int k0 = 0; k0 < 32; k0 += 4) {
      v2f A0 = frag_at(lM, 128, m0, k0, lane);
      v2f A1 = frag_at(lM, 128, m0 + 16, k0, lane);
      v2f B0 = frag_b(lB, 32, k0, 0, lane);
      v2f B1 = frag_b(lB, 32, k0, 16, lane);
      a00 = wmma4(A0, B0, a00);
      a01 = wmma4(A0, B1, a01);
      a10 = wmma4(A1, B0, a10);
      a11 = wmma4(A1, B1, a11);
    }
    store_d(lU, 32, m0, 0, lane, a00);
    store_d(lU, 32, m0, 16, lane, a01);
    store_d(lU, 32, m0 + 16, 0, lane, a10);
    store_d(lU, 32, m0 + 16, 16, lane, a11);
  }
  __syncthreads();

  // --- Y = U * M_top, scaled; wave owns rows m0..m0+31, loop column pairs
  float* yp = out + (size_t)bo * 16384;
  const int m0 = wave * 32;
  const float s = 1.0f / 16384.0f;
  for (int np = 0; np < 4; ++np) {
    const int n0 = np * 32;
    v8f a00 = {}, a01 = {}, a10 = {}, a11 = {};
    for (int k0 = 0; k0 < 32; k0 += 4) {
      v2f A0 = frag_a(lU, 32, m0, k0, lane);
      v2f A1 = frag_a(lU, 32, m0 + 16, k0, lane);
      v2f B0 = frag_b(lM, 128, k0, n0, lane);
      v2f B1 = frag_b(lM, 128, k0, n0 + 16, lane);
      a00 = wmma4(A0, B0, a00);
      a01 = wmma4(A0, B1, a01);
      a10 = wmma4(A1, B0, a10);
      a11 = wmma4(A1, B1, a11);
    }
    const int mm = m0 + ((lane >> 4) << 3);
    const int nn = n0 + (lane & 15);
#pragma unroll
    for (int r = 0; r < 8; ++r) {
      yp[(size_t)(mm + r) * 128 + nn] = s * a00[r];
      yp[(size_t)(mm + r) * 128 + nn + 16] = s * a01[r];
      yp[(size_t)(mm + 16 + r) * 128 + nn] = s * a10[r];
      yp[(size_t)(mm + 16 + r) * 128 + nn + 16] = s * a11[r];
    }
  }
}

// ---------------------------------------------------------------------------
extern "C" void kernel_launch(void* const* d_in, const int* in_sizes, int n_in,
                              void* d_out, int out_size, void* d_ws,
                              size_t ws_size, hipStream_t stream) {
  const float* x = (const float*)d_in[0];        // [32,64,128,128]
  const float* w = (const float*)d_in[1];        // [64,64,32,32]
  float* out = (float*)d_out;                    // [32,64,128,128]

  float* M = (float*)d_ws;                       // 16384 floats
  float* Xt = M + 16384;                         // 1024*2048 floats
  float* Blk = Xt + 1024 * 2048;                 // 1024*2048 floats  (~17 MB total)

  build_M_kernel<<<64, 256, 0, stream>>>(M);
  fwd_kernel<<<2048, 128, 0, stream>>>(x, M, Xt);
  mix_kernel<<<1024, 256, 0, stream>>>(Xt, w, Blk);
  inv_kernel<<<2048, 128, 0, stream>>>(Blk, M, out);
}